// LabelSmoothing_3848290697270
// MI455X (gfx1250) — compile-verified
//
#include <hip/hip_runtime.h>

// ---------------------------------------------------------------------------
// Label-smoothing KL loss, collapsed analytically:
//   per valid row:  C - ( u*(rowsum - p[0] - p[t]) + conf*p[t] )
//   C = (V-2)*u*log(u) + conf*log(conf),  u = 0.1/(V-2), conf = 0.9
// Stage 1: bandwidth-bound row sums (256 MiB streamed once, NT hint).
//          Grid is padded to a multiple of 64 rows; pad rows write 0.0 so
//          stage 2 needs no bounds checks (keeps EXEC all-ones, no masking).
// Stage 2: single-wave WMMA f32 16x16x4 reduction of the padded partials.
// ---------------------------------------------------------------------------

typedef float v4f __attribute__((ext_vector_type(4)));
typedef float v2f __attribute__((ext_vector_type(2)));
typedef float v8f __attribute__((ext_vector_type(8)));

#define SMOOTHING 0.1f
#define CONF      0.9f

// -------------------- Stage 1: per-row streaming reduction ------------------
__global__ __launch_bounds__(256)
void ls_row_reduce(const float* __restrict__ pred,
                   const int*   __restrict__ tgt,
                   float*       __restrict__ partial,
                   int vocab, int nrows)
{
    const int row = blockIdx.x;
    const int tid = threadIdx.x;

    // Padding rows: contribute exactly zero, touch no input memory.
    if (row >= nrows) {
        if (tid == 0) partial[row] = 0.0f;
        return;
    }

    const float* __restrict__ p = pred + (size_t)row * (size_t)vocab;

    // Coalesced non-temporal float4 streaming (single pass, dataset > L2).
    const v4f* __restrict__ vp = reinterpret_cast<const v4f*>(p);
    const int nvec = vocab >> 2;   // vocab is a multiple of 4 (32000)

    float a0 = 0.f, a1 = 0.f, a2 = 0.f, a3 = 0.f;
    for (int i = tid; i < nvec; i += 256) {
        v4f v = __builtin_nontemporal_load(vp + i);
        a0 += v.x; a1 += v.y; a2 += v.z; a3 += v.w;
    }
    float s = (a0 + a1) + (a2 + a3);

    // Deterministic block tree reduction.
    __shared__ float red[256];
    red[tid] = s;
    __syncthreads();
    #pragma unroll
    for (int off = 128; off > 0; off >>= 1) {
        if (tid < off) red[tid] += red[tid + off];
        __syncthreads();
    }

    if (tid == 0) {
        const int t = tgt[row];
        float out = 0.0f;
        if (t != 0) {
            const float vm2  = (float)(vocab - 2);
            const float u    = SMOOTHING / vm2;
            // xlogy constant term (closed form, f32-safe magnitudes)
            const float C    = vm2 * u * logf(u) + CONF * logf(CONF);
            const float p0   = p[0];
            const float pt   = p[t];
            const float rest = red[0] - p0 - pt;
            out = C - (u * rest + CONF * pt);
        }
        partial[row] = out;
    }
}

// -------------------- Stage 2: WMMA-based final reduction -------------------
#if __has_builtin(__builtin_amdgcn_wmma_f32_16x16x4_f32)
#define LS_HAVE_WMMA_F32 1
#endif

__global__ __launch_bounds__(32)
void ls_final_reduce(const float* __restrict__ partial,
                     float*       __restrict__ out,
                     int npad)    // multiple of 64, guaranteed by launcher
{
    const int lane = threadIdx.x;   // exactly one wave32, EXEC all ones

#if LS_HAVE_WMMA_F32
    // D = A x B + C with B = ones(4x16): every D[m][n] accumulates the
    // row-sum of A's 16x4 tile. The element->(m,k) mapping is irrelevant
    // because we sum all entries, so each lane grabs 2 consecutive partials
    // per 64-element chunk (one global_load_b64, no predication).
    v8f c0 = {};
    v8f c1 = {};
    v2f ones; ones.x = 1.0f; ones.y = 1.0f;

    const v2f* __restrict__ vp = reinterpret_cast<const v2f*>(partial);
    const int nchunks = npad >> 6;           // 64 values per chunk

    int ch = 0;
    // Two independent accumulators -> two WMMAs in flight per trip.
    for (; ch + 2 <= nchunks; ch += 2) {
        v2f a0 = vp[(ch << 5) + lane];
        v2f a1 = vp[((ch + 1) << 5) + lane];
        c0 = __builtin_amdgcn_wmma_f32_16x16x4_f32(
                 false, a0, false, ones, (short)0, c0, false, false);
        c1 = __builtin_amdgcn_wmma_f32_16x16x4_f32(
                 false, a1, false, ones, (short)0, c1, false, false);
    }
    for (; ch < nchunks; ++ch) {
        v2f a0 = vp[(ch << 5) + lane];
        c0 = __builtin_amdgcn_wmma_f32_16x16x4_f32(
                 false, a0, false, ones, (short)0, c0, false, false);
    }

    // C/D layout: column n lives in lane n (M=0..7 in v[0..7]) and lane n+16
    // (M=8..15). Take column 0: total = colsum(lane0) + colsum(lane16).
    v8f c;
    #pragma unroll
    for (int i = 0; i < 8; ++i) c[i] = c0[i] + c1[i];
    float s = ((c[0] + c[1]) + (c[2] + c[3])) + ((c[4] + c[5]) + (c[6] + c[7]));
    float total = __shfl(s, 0, 32) + __shfl(s, 16, 32);
    if (lane == 0) out[0] = total;
#else
    // Fallback: deterministic lane-strided + sequential combine.
    float s = 0.f;
    for (int i = lane; i < npad; i += 32) s += partial[i];
    __shared__ float red[32];
    red[lane] = s;
    __syncthreads();
    if (lane == 0) {
        float tot = 0.f;
        for (int i = 0; i < 32; ++i) tot += red[i];
        out[0] = tot;
    }
#endif
}

// ---------------------------------------------------------------------------
extern "C" void kernel_launch(void* const* d_in, const int* in_sizes, int n_in,
                              void* d_out, int out_size, void* d_ws, size_t ws_size,
                              hipStream_t stream)
{
    const float* pred = (const float*)d_in[0];   // (B*S, V) log-probs, f32
    const int*   tgt  = (const int*)d_in[1];     // (B*S,) targets
    float*       out  = (float*)d_out;           // scalar
    float*       ws   = (float*)d_ws;            // per-row partials (padded)

    const int nrows = in_sizes[1];               // 2048
    const int vocab = in_sizes[0] / nrows;       // 32000
    const int npad  = (nrows + 63) & ~63;        // multiple of 64 for stage 2

    ls_row_reduce<<<npad, 256, 0, stream>>>(pred, tgt, ws, vocab, nrows);
    ls_final_reduce<<<1, 32, 0, stream>>>(ws, out, npad);
}